// Pollu_Model_28793460752828
// MI455X (gfx1250) — compile-verified
//
#include <hip/hip_runtime.h>

typedef __attribute__((ext_vector_type(2))) float v2f;
typedef __attribute__((ext_vector_type(8))) float v8f;

// POLLU reaction gather indices (0-based reactions, padded to 28; idx 20 == constant-1 row)
static constexpr int ONES = 20;
static constexpr int IAI[28] = {0,1,4,6,6,6,8,8,10,10,12,9,13,0,2,3,3,15,15,16,18,18,0,18,19, ONES,ONES,ONES};
static constexpr int IBI[28] = {ONES,3,1,ONES,ONES,5,ONES,5,1,0,ONES,1,ONES,5,ONES,ONES,ONES,ONES,ONES,5,ONES,ONES,3,0,ONES, ONES,ONES,ONES};

// Stoichiometry S[species][reaction], rows 0..19 = species, 16..19 duplicated at rows 16..19
// for the second WMMA chain (rows 20..31 zero). 28 columns (25 reactions + 3 zero pad).
__constant__ float S_TAB[32][28] = {
  /* sp0  */ {-1,1,1,0,0,0,0,0,1,-1,1,1,0,-1,0,0,0,0,0,0,0,1,-1,-1,1,0,0,0},
  /* sp1  */ {1,-1,-1,0,0,0,0,0,-1,0,0,-1,0,0,0,0,0,0,0,0,1,0,0,0,0,0,0,0},
  /* sp2  */ {1,0,0,0,0,0,0,0,0,0,0,0,0,0,-1,0,1,0,1,0,0,1,0,0,0,0,0,0},
  /* sp3  */ {0,-1,0,0,0,0,0,0,0,0,0,0,0,0,1,-1,-1,0,0,0,0,0,-1,0,0,0,0,0},
  /* sp4  */ {0,0,-1,2,0,1,1,0,0,0,0,0,1,0,0,0,0,0,0,1,0,0,0,0,0,0,0,0},
  /* sp5  */ {0,0,1,0,0,-1,0,-1,0,0,0,0,0,-1,0,0,0,2,0,-1,0,0,0,0,0,0,0,0},
  /* sp6  */ {0,0,0,-1,-1,-1,0,0,0,0,0,0,1,0,0,0,0,0,0,0,0,0,0,0,0,0,0,0},
  /* sp7  */ {0,0,0,1,1,1,1,0,0,0,0,0,0,0,0,0,0,0,0,0,0,0,0,0,0,0,0,0},
  /* sp8  */ {0,0,0,0,0,0,-1,-1,0,0,0,0,0,0,0,0,0,0,0,0,0,0,0,0,0,0,0,0},
  /* sp9  */ {0,0,0,0,0,0,1,0,1,0,0,-1,0,0,0,0,0,0,0,0,0,0,0,0,0,0,0,0},
  /* sp10 */ {0,0,0,0,0,0,0,1,-1,-1,1,0,0,0,0,0,0,0,0,0,0,0,0,0,0,0,0,0},
  /* sp11 */ {0,0,0,0,0,0,0,0,1,0,0,0,0,0,0,0,0,0,0,0,0,0,0,0,0,0,0,0},
  /* sp12 */ {0,0,0,0,0,0,0,0,0,1,-1,0,0,0,0,0,0,0,0,0,0,0,0,0,0,0,0,0},
  /* sp13 */ {0,0,0,0,0,0,0,0,0,0,0,1,-1,0,0,0,0,0,0,0,0,0,0,0,0,0,0,0},
  /* sp14 */ {0,0,0,0,0,0,0,0,0,0,0,0,0,1,0,0,0,0,0,0,0,0,0,0,0,0,0,0},
  /* sp15 */ {0,0,0,0,0,0,0,0,0,0,0,0,0,0,0,1,0,-1,-1,0,0,0,0,0,0,0,0,0},
  /* sp16 */ {0,0,0,0,0,0,0,0,0,0,0,0,0,0,0,0,0,0,0,-1,0,0,0,0,0,0,0,0},
  /* sp17 */ {0,0,0,0,0,0,0,0,0,0,0,0,0,0,0,0,0,0,0,1,0,0,0,0,0,0,0,0},
  /* sp18 */ {0,0,0,0,0,0,0,0,0,0,0,0,0,0,0,0,0,0,0,0,-1,-1,1,-1,1,0,0,0},
  /* sp19 */ {0,0,0,0,0,0,0,0,0,0,0,0,0,0,0,0,0,0,0,0,0,0,0,1,-1,0,0,0},
  {0},{0},{0},{0},{0},{0},{0},{0},{0},{0},{0},{0}
};

__device__ __forceinline__ float ld_c(const float* __restrict__ conc, long Bn, long elem, int idx) {
  // idx is a compile-time constant after full unroll; the ONES row folds to 1.0f (no load).
  return (idx == ONES) ? 1.0f : conc[(long)idx * Bn + elem];
}

__global__ __launch_bounds__(256) void pollu_wmma_kernel(const float* __restrict__ conc,
                                                         const float* __restrict__ kptr,
                                                         float* __restrict__ out,
                                                         int Bn) {
  const int lane = threadIdx.x & 31;
  const int wave = threadIdx.x >> 5;
  const long tile = (long)blockIdx.x * 8 + wave;   // one 16-element tile per wave
  if (tile * 16 >= Bn) return;                     // wave-uniform: EXEC stays all-1s for WMMA
  const int  m    = lane & 15;                     // species row within half / element within tile
  const bool lo   = lane < 16;
  const long elemR = tile * 16 + m;                // real element
  const long elem  = (elemR < Bn) ? elemR : (Bn - 1); // clamped for loads (tail safety)

  // Rate constants: uniform scalar loads; pads fold to 0 so pad fluxes vanish.
  float kk[28];
  #pragma unroll
  for (int r = 0; r < 28; ++r) kk[r] = (r < 25) ? kptr[r] : 0.0f;

  // ---- flux directly in B-matrix layout (4x16 per chunk) ----
  // Assumed f32 B layout mirrors the documented A layout: VGPR v, lanes0-15 -> K=4c+v,
  // lanes16-31 -> K=4c+2+v (N = lane&15 across lanes).
  v2f bm[7];
  #pragma unroll
  for (int c = 0; c < 7; ++c) {
    float f0, f1;
    if (lo) {
      f0 = kk[4*c+0] * ld_c(conc, Bn, elem, IAI[4*c+0]) * ld_c(conc, Bn, elem, IBI[4*c+0]);
      f1 = kk[4*c+1] * ld_c(conc, Bn, elem, IAI[4*c+1]) * ld_c(conc, Bn, elem, IBI[4*c+1]);
    } else {
      f0 = kk[4*c+2] * ld_c(conc, Bn, elem, IAI[4*c+2]) * ld_c(conc, Bn, elem, IBI[4*c+2]);
      f1 = kk[4*c+3] * ld_c(conc, Bn, elem, IAI[4*c+3]) * ld_c(conc, Bn, elem, IBI[4*c+3]);
    }
    bm[c].x = f0; bm[c].y = f1;
  }

  // ---- A matrices: constant stoichiometry chunks (16x4), layout per ISA:
  // lanes0-15: M=lane, VGPR0=K(4c), VGPR1=K(4c+1); lanes16-31: M=lane-16, K(4c+2), K(4c+3).
  const int kof = lo ? 0 : 2;
  v2f am[7], am2[3];
  #pragma unroll
  for (int c = 0; c < 7; ++c) {
    am[c].x = S_TAB[m][4*c + kof];
    am[c].y = S_TAB[m][4*c + kof + 1];
  }
  #pragma unroll
  for (int c = 0; c < 3; ++c) {            // species 16..19 only touch K-chunks 4,5,6
    am2[c].x = S_TAB[16 + m][4*(4+c) + kof];
    am2[c].y = S_TAB[16 + m][4*(4+c) + kof + 1];
  }

  // ---- D = S_chunk x flux_chunk accumulation on the matrix pipe ----
  v8f acc = {0.f,0.f,0.f,0.f,0.f,0.f,0.f,0.f};
  #pragma unroll
  for (int c = 0; c < 7; ++c)
    acc = __builtin_amdgcn_wmma_f32_16x16x4_f32(false, am[c], false, bm[c],
                                                (short)0, acc, false, false);
  v8f acc2 = {0.f,0.f,0.f,0.f,0.f,0.f,0.f,0.f};
  #pragma unroll
  for (int c = 0; c < 3; ++c)
    acc2 = __builtin_amdgcn_wmma_f32_16x16x4_f32(false, am2[c], false, bm[4 + c],
                                                 (short)0, acc2, false, false);

  // ---- stores: D layout VGPR j = species (j | j+8) for element (lane&15) ----
  if (elemR < Bn) {
    float* o1 = out + elemR * 20 + (lo ? 0 : 8);
    #pragma unroll
    for (int j = 0; j < 8; ++j) o1[j] = acc[j];
    if (lo) {                               // species 16..19 live in lanes 0-15 of chain 2
      float* o2 = out + elemR * 20 + 16;
      #pragma unroll
      for (int j = 0; j < 4; ++j) o2[j] = acc2[j];
    }
  }
}

extern "C" void kernel_launch(void* const* d_in, const int* in_sizes, int n_in,
                              void* d_out, int out_size, void* d_ws, size_t ws_size,
                              hipStream_t stream) {
  // inputs: d_in[0]=t (unused), d_in[1]=conc_in [20,B] f32, d_in[2]=k [25] f32
  const float* conc = (const float*)d_in[1];
  const float* k    = (const float*)d_in[2];
  float* out        = (float*)d_out;
  const int Bn      = in_sizes[1] / 20;     // 2,000,000
  const int grid    = (Bn + 127) / 128;     // 128 elements per 256-thread block (8 waves x 16)
  pollu_wmma_kernel<<<grid, 256, 0, stream>>>(conc, k, out, Bn);
}